// DiscreteContinuousConv2d_89507118449171
// MI455X (gfx1250) — compile-verified
//
#include <hip/hip_runtime.h>

typedef __attribute__((ext_vector_type(2))) float v2f;
typedef __attribute__((ext_vector_type(8))) float v8f;

#define N_PTS  16384      // n_in == n_out (128x128 grid)
#define BATCH  4
#define CIN    64
#define COUT   64
#define KSZ    9
#define BC     (BATCH * CIN)   // 256 floats per grid point

// ---------------------------------------------------------------------------
// Phase 0: xt[m*256 + b*64 + c] = x[b][c][m] * quad[m]
// Coalesced writes; gather reads are L2-friendly (x is only 16 MB).
// ---------------------------------------------------------------------------
__global__ void xq_transpose_kernel(const float* __restrict__ x,
                                    const float* __restrict__ quad,
                                    float* __restrict__ xt) {
  int idx = blockIdx.x * blockDim.x + threadIdx.x;   // 0 .. N_PTS*BC-1
  int m = idx >> 8;                                  // grid point
  int j = idx & 255;                                 // b*64 + c
  int b = j >> 6, c = j & 63;
  xt[idx] = x[(b * CIN + c) * N_PTS + m] * quad[m];
}

__global__ void zero_kernel(float* __restrict__ p, int n) {
  int i = blockIdx.x * blockDim.x + threadIdx.x;
  if (i < n) p[i] = 0.0f;
}

// ---------------------------------------------------------------------------
// Phase 1 (per k, per b): Tk[m][b*64+o] = sum_c xt[m][b*64+c] * w[o][c][k]
// GEMM M=16384, K=64, N=64 using V_WMMA_F32_16X16X4_F32.
// Block = 256 threads = 8 waves; block tile = 32 rows x 64 cols;
// each wave owns one 16x16 output tile, 16 WMMA steps over K.
// EXEC is all-1s throughout (exact grid, no divergence) per WMMA rules.
// ---------------------------------------------------------------------------
__global__ void gemm_wmma_kernel(const float* __restrict__ xt,
                                 const float* __restrict__ weight,
                                 float* __restrict__ Tk, int k) {
  __shared__ float a_lds[32 * 66];   // padded stride 66 to dodge bank conflicts
  __shared__ float w_lds[64 * 66];   // w_lds[o][c] = weight[o][c][k]

  const int tid = threadIdx.x;
  const int b   = blockIdx.y;
  const int m0  = blockIdx.x * 32;

  // Stage B (weights) transposed: w_lds[o*66 + c]
  for (int i = tid; i < 64 * 64; i += 256) {
    int o = i >> 6, c = i & 63;
    w_lds[o * 66 + c] = weight[(o * CIN + c) * KSZ + k];
  }
  // Stage A tile: a_lds[r*66 + c] = xt[(m0+r)*256 + b*64 + c] (coalesced)
  for (int i = tid; i < 32 * 64; i += 256) {
    int r = i >> 6, c = i & 63;
    a_lds[r * 66 + c] = xt[(m0 + r) * BC + b * CIN + c];
  }
  __syncthreads();

  const int lane = tid & 31;
  const int hi   = lane >> 4;        // K-half select (ISA 16x4 f32 A layout)
  const int l15  = lane & 15;
  const int wave = tid >> 5;
  const int wm   = wave >> 2;        // 0..1 : M sub-tile
  const int wn   = wave & 3;         // 0..3 : N sub-tile

  v8f acc = {};
#pragma unroll
  for (int ks = 0; ks < 16; ++ks) {
    const int kk = ks * 4 + hi * 2;
    // A 16x4: lanes 0-15 -> K={kk,kk+1}, lanes 16-31 -> K={kk+2,kk+3}, M=lane&15
    v2f a  = { a_lds[(wm * 16 + l15) * 66 + kk],
               a_lds[(wm * 16 + l15) * 66 + kk + 1] };
    // B 4x16: same K striping, N = lane&15
    v2f bw = { w_lds[(wn * 16 + l15) * 66 + kk],
               w_lds[(wn * 16 + l15) * 66 + kk + 1] };
    acc = __builtin_amdgcn_wmma_f32_16x16x4_f32(
        /*neg_a=*/false, a, /*neg_b=*/false, bw,
        /*c_mod=*/(short)0, acc, /*reuse_a=*/false, /*reuse_b=*/false);
  }

  // C layout: VGPR r -> M = r (lanes 0-15) / M = 8+r (lanes 16-31), N = lane&15
  const int o     = wn * 16 + l15;
  const int mbase = m0 + wm * 16 + hi * 8;
#pragma unroll
  for (int r = 0; r < 8; ++r) {
    Tk[(mbase + r) * BC + b * CIN + o] = acc[r];
  }
}

// ---------------------------------------------------------------------------
// Phase 2 (per k): one wave per sparse entry with matching kernel index.
// Gather 256 contiguous floats from Tk[in], scale by psi, atomic-add 256
// contiguous floats into ytmp[out]. Both buffers are L2-resident (16.7 MB).
// ---------------------------------------------------------------------------
__global__ void scatter_kernel(const int* __restrict__ kI,
                               const int* __restrict__ oI,
                               const int* __restrict__ iI,
                               const float* __restrict__ psi,
                               const float* __restrict__ Tk,
                               float* __restrict__ ytmp,
                               int k, int nnz) {
  int gid  = blockIdx.x * blockDim.x + threadIdx.x;
  int e    = gid >> 5;
  int lane = gid & 31;
  if (e >= nnz) return;
  if (kI[e] != k) return;

  float p  = psi[e];
  int outp = oI[e];
  int inp  = iI[e];

  const float4* src = (const float4*)(Tk + (size_t)inp * BC) + lane * 2;
  float4 v0 = src[0];
  float4 v1 = src[1];

  float* dst = ytmp + (size_t)outp * BC + lane * 8;
  atomicAdd(dst + 0, p * v0.x);
  atomicAdd(dst + 1, p * v0.y);
  atomicAdd(dst + 2, p * v0.z);
  atomicAdd(dst + 3, p * v0.w);
  atomicAdd(dst + 4, p * v1.x);
  atomicAdd(dst + 5, p * v1.y);
  atomicAdd(dst + 6, p * v1.z);
  atomicAdd(dst + 7, p * v1.w);
}

// ---------------------------------------------------------------------------
// Phase 3: y[b][o][n] = ytmp[n][b*64+o] + bias[o]   (coalesced writes)
// ---------------------------------------------------------------------------
__global__ void finalize_kernel(const float* __restrict__ ytmp,
                                const float* __restrict__ bias,
                                float* __restrict__ y) {
  int idx = blockIdx.x * blockDim.x + threadIdx.x;   // (b*COUT+o)*N_PTS + n
  int n  = idx & (N_PTS - 1);
  int bo = idx >> 14;
  int o  = bo & 63;
  int b  = bo >> 6;
  y[idx] = ytmp[(size_t)n * BC + b * CIN + o] + bias[o];
}

// ---------------------------------------------------------------------------
extern "C" void kernel_launch(void* const* d_in, const int* in_sizes, int n_in,
                              void* d_out, int out_size, void* d_ws, size_t ws_size,
                              hipStream_t stream) {
  const float* x        = (const float*)d_in[0];   // (B, Cin, n_in)
  const int*   psi_idx  = (const int*)  d_in[1];   // (3, nnz)
  const float* psi_vals = (const float*)d_in[2];   // (nnz,)
  const float* quad     = (const float*)d_in[3];   // (n_in,)
  const float* weight   = (const float*)d_in[4];   // (Cout, Cin, K)
  const float* bias     = (const float*)d_in[5];   // (Cout,)
  float*       y        = (float*)d_out;

  const int nnz = in_sizes[2];

  // Workspace layout: xt | Tk | ytmp  (each N_PTS*BC floats = 16.7 MB)
  float* xt   = (float*)d_ws;
  float* Tk   = xt + (size_t)N_PTS * BC;
  float* ytmp = Tk + (size_t)N_PTS * BC;

  const int* kI = psi_idx;
  const int* oI = psi_idx + nnz;
  const int* iI = psi_idx + 2 * nnz;

  const int elems = N_PTS * BC;  // 4,194,304

  xq_transpose_kernel<<<elems / 256, 256, 0, stream>>>(x, quad, xt);
  zero_kernel<<<elems / 256, 256, 0, stream>>>(ytmp, elems);

  const int sblocks = (nnz * 32 + 255) / 256;
  for (int k = 0; k < KSZ; ++k) {
    gemm_wmma_kernel<<<dim3(N_PTS / 32, BATCH), 256, 0, stream>>>(xt, weight, Tk, k);
    scatter_kernel<<<sblocks, 256, 0, stream>>>(kI, oI, iI, psi_vals, Tk, ytmp, k, nnz);
  }

  finalize_kernel<<<(BATCH * COUT * N_PTS) / 256, 256, 0, stream>>>(ytmp, bias, y);
}